// plane_rotations_24653112279145
// MI455X (gfx1250) — compile-verified
//
#include <hip/hip_runtime.h>

typedef float v2f __attribute__((ext_vector_type(2)));
typedef float v8f __attribute__((ext_vector_type(8)));
typedef unsigned int u32;
typedef u32 u32x4 __attribute__((ext_vector_type(4)));
typedef int  i32x8 __attribute__((ext_vector_type(8)));

#define NDIM   256
#define NCHUNK 16

// ---------------------------------------------------------------------------
// Kernel 1: build per-chunk Givens products.
// Grid: NCHUNK chunks x 4 column groups. Block: 64 threads (2 waves), each
// thread exclusively owns one matrix column in a 256x64 LDS slab (64 KB).
// Rotations only mix rows i,j, so columns never interact -> no barriers.
// Runs of equal idx_i keep row i in a register (1 LDS load + 1 store / rot).
// Output: Q_k = P_k^T stored row-major (256x256) so the combine tree computes
// M^T = Q_1 @ Q_2 @ ... @ Q_16 directly.
// ---------------------------------------------------------------------------
__global__ __launch_bounds__(64) void build_chunks(
    const float* __restrict__ rots, const int* __restrict__ idx_i,
    const int* __restrict__ idx_j, float* __restrict__ qout, int nrot)
{
  __shared__ float sM[NDIM * 64];            // 64 KB: [row][ownedCol]
  const int tid   = threadIdx.x;             // 0..63
  const int chunk = blockIdx.x >> 2;
  const int cg    = blockIdx.x & 3;
  const int c     = cg * 64 + tid;           // global column owned by thread

  for (int r = 0; r < NDIM; ++r)
    sM[r * 64 + tid] = (r == c) ? 1.0f : 0.0f;

  int k        = (chunk * nrot) / NCHUNK;
  const int k1 = ((chunk + 1) * nrot) / NCHUNK;
  while (k < k1) {
    const int i = idx_i[k];
    float acc = sM[i * 64 + tid];            // row i held in a register
    do {
      const int   j   = idx_j[k];
      const float th  = rots[k];
      const float cth = __cosf(th);
      const float sth = __sinf(th);
      const float rj  = sM[j * 64 + tid];
      const float na  = cth * acc - sth * rj;
      sM[j * 64 + tid] = fmaf(sth, acc, cth * rj);
      acc = na;
      ++k;
    } while (k < k1 && idx_i[k] == i);
    sM[i * 64 + tid] = acc;
  }

  // store transposed: Q[c][r] = P[r][c]
  float* q = qout + (size_t)chunk * (NDIM * NDIM);
  for (int r = 0; r < NDIM; ++r)
    q[c * NDIM + r] = sM[r * 64 + tid];
}

// ---------------------------------------------------------------------------
// Kernel 2: pairwise 256x256x256 f32 WMMA matmul for the combine tree.
// One wave per 16x16 output tile; D[prod] = src[2*prod] @ src[2*prod+1].
// ---------------------------------------------------------------------------
__global__ __launch_bounds__(32) void combine_pairs(
    const float* __restrict__ src, float* __restrict__ dst)
{
  const int prod = blockIdx.x >> 8;
  const int tile = blockIdx.x & 255;
  const int tm = tile >> 4, tn = tile & 15;
  const int lane = threadIdx.x;
  const int l15  = lane & 15;
  const int kh   = lane >> 4;                // 0 or 1

  const float* A = src + (size_t)(2 * prod)     * (NDIM * NDIM);
  const float* B = src + (size_t)(2 * prod + 1) * (NDIM * NDIM);
  float*       D = dst + (size_t)prod           * (NDIM * NDIM);

  v8f acc = {};
  const int arow = tm * 16 + l15;
  for (int ks = 0; ks < NDIM / 4; ++ks) {
    const int kb = ks * 4 + kh * 2;
    v2f a = *(const v2f*)(A + arow * NDIM + kb);
    v2f b;
    b.x = B[(kb + 0) * NDIM + tn * 16 + l15];
    b.y = B[(kb + 1) * NDIM + tn * 16 + l15];
    acc = __builtin_amdgcn_wmma_f32_16x16x4_f32(
        false, a, false, b, (short)0, acc, false, false);
  }
#pragma unroll
  for (int v = 0; v < 8; ++v)
    D[(tm * 16 + v + 8 * kh) * NDIM + tn * 16 + l15] = acc[v];
}

// ---------------------------------------------------------------------------
// Kernel 3: out = x @ W   (x: nrows x 256 f32, W = M^T row-major 256x256).
// Block = 256 threads (8 waves) computes 128 rows x 256 cols. W is staged in
// four 64x256 f32 LDS panels (64 KB) moved by the Tensor Data Mover: wave 0
// issues a single tensor_load_to_lds per panel (D# per CDNA5 ISA, 2D tile:
// tile_dim0=256 cols, tile_dim1=64 rows, stride 256, data_size=4B), waits on
// TENSORcnt, and the block barrier releases the other waves. Each wave owns a
// 16x256 accumulator strip (16 tiles x v8f = 128 VGPRs). x is read once.
// ---------------------------------------------------------------------------
__global__ __launch_bounds__(256) void apply_gemm(
    const float* __restrict__ x, const float* __restrict__ W,
    float* __restrict__ out)
{
  __shared__ float sW[64 * NDIM];            // 64 KB panel: [kInPanel][n]
  const int tid  = threadIdx.x;
  const int lane = tid & 31;
  const int wave = tid >> 5;                 // 0..7
  const int l15  = lane & 15;
  const int kh   = lane >> 4;
  const int rowBase = blockIdx.x * 128 + wave * 16;

  v8f acc[16] = {};

  for (int p = 0; p < 4; ++p) {
    // --- TDM: DMA the 64x256 f32 panel W[p*64 .. p*64+64) into LDS ---------
    if (wave == 0) {
      const unsigned long long ga =
          (unsigned long long)(const void*)(W + (size_t)p * 64 * NDIM);
      const u32 ldsAddr = (u32)(size_t)(&sW[0]);   // LDS aperture low bits

      u32x4 g0;
      g0.x = 1u;                                   // count=1, load descriptor
      g0.y = ldsAddr;                              // D#.lds_addr
      g0.z = (u32)(ga & 0xffffffffu);              // global_addr[31:0]
      g0.w = (u32)((ga >> 32) & 0x01ffffffu)       // global_addr[56:32]
           | (2u << 30);                           // type = 2 (image)

      i32x8 g1;
      g1[0] = (int)(2u << 16);                     // data_size = 2 -> 4 bytes
      g1[1] = (int)((u32)NDIM << 16);              // tensor_dim0[15:0] = 256
      g1[2] = (int)((u32)NDIM << 16);              // tensor_dim1[15:0] = 256
      g1[3] = (int)((u32)NDIM << 16);              // tile_dim0      = 256
      g1[4] = 64;                                  // tile_dim1=64, tile_dim2=0
      g1[5] = NDIM;                                // tensor_dim0_stride = 256
      g1[6] = 0;
      g1[7] = 0;

      asm volatile("tensor_load_to_lds %0, %1"
                   :: "s"(g0), "s"(g1)
                   : "memory");
      __builtin_amdgcn_s_wait_tensorcnt(0);
    }
    __syncthreads();                               // panel visible to all waves

#pragma unroll 4
    for (int ks = 0; ks < 16; ++ks) {
      const int kb = ks * 4 + kh * 2;        // k within panel
      v2f a = *(const v2f*)(x + (size_t)(rowBase + l15) * NDIM + p * 64 + kb);
#pragma unroll
      for (int nt = 0; nt < 16; ++nt) {
        v2f b;
        b.x = sW[(kb + 0) * NDIM + nt * 16 + l15];
        b.y = sW[(kb + 1) * NDIM + nt * 16 + l15];
        acc[nt] = __builtin_amdgcn_wmma_f32_16x16x4_f32(
            false, a, false, b, (short)0, acc[nt], false, false);
      }
    }
    __syncthreads();                               // done reading before next DMA
  }

#pragma unroll
  for (int nt = 0; nt < 16; ++nt) {
#pragma unroll
    for (int v = 0; v < 8; ++v)
      out[(size_t)(rowBase + v + 8 * kh) * NDIM + nt * 16 + l15] = acc[nt][v];
  }
}

// ---------------------------------------------------------------------------
// Host launcher. Workspace: bufA = 16 matrices, bufB = 8 matrices (6 MB).
// Tree: 16 -> 8 -> 4 -> 2 -> 1; final W lands back in bufA.
// ---------------------------------------------------------------------------
extern "C" void kernel_launch(void* const* d_in, const int* in_sizes, int n_in,
                              void* d_out, int out_size, void* d_ws, size_t ws_size,
                              hipStream_t stream) {
  const float* x    = (const float*)d_in[0];
  const float* rots = (const float*)d_in[1];
  const int*   ii   = (const int*)d_in[2];
  const int*   jj   = (const int*)d_in[3];
  float* out = (float*)d_out;
  const int nrot  = in_sizes[1];
  const int nrows = in_sizes[0] / NDIM;      // 65536

  float* bufA = (float*)d_ws;
  float* bufB = bufA + (size_t)NCHUNK * NDIM * NDIM;

  build_chunks<<<NCHUNK * 4, 64, 0, stream>>>(rots, ii, jj, bufA, nrot);

  combine_pairs<<<8 * 256, 32, 0, stream>>>(bufA, bufB);
  combine_pairs<<<4 * 256, 32, 0, stream>>>(bufB, bufA);
  combine_pairs<<<2 * 256, 32, 0, stream>>>(bufA, bufB);
  combine_pairs<<<1 * 256, 32, 0, stream>>>(bufB, bufA);

  apply_gemm<<<nrows / 128, 256, 0, stream>>>(x, bufA, out);
}